// GlobalAttention_75076028334716
// MI455X (gfx1250) — compile-verified
//
#include <hip/hip_runtime.h>
#include <math.h>

// Problem constants (fixed by the reference): N=1024 nodes, D=attn=256.
#define NN   1024
#define DD   256
#define NEG_SLOPE 0.2f

typedef float v2f __attribute__((ext_vector_type(2)));
typedef float v4f __attribute__((ext_vector_type(4)));
typedef float v8f __attribute__((ext_vector_type(8)));

__device__ __forceinline__ float lrelu(float x) {
    return fmaxf(x, 0.0f) + NEG_SLOPE * fminf(x, 0.0f);
}

// ---------------------------------------------------------------------------
// Kernel 1: Wh1 = H @ W1^T, Wh2 = H @ W2^T via V_WMMA_F32_16X16X4_F32.
// One wave computes a 16x16 f32 tile; 64 K-steps of 4. grid.z selects W1/W2.
// A 16x4 f32 layout: lanes 0-15 -> M=lane, VGPR0/1 = K0/K1; lanes 16-31 ->
// M=lane-16, K2/K3. K contiguous in both H and W rows -> single v2f load.
// ---------------------------------------------------------------------------
__global__ void __launch_bounds__(32)
k_gemm_wh(const float* __restrict__ H, const float* __restrict__ W,
          float* __restrict__ Wh1, float* __restrict__ Wh2) {
    const int lane = threadIdx.x;
    const int half = lane >> 4;
    const int l16  = lane & 15;
    const int row0 = blockIdx.x * 16;
    const int col0 = blockIdx.y * 16;
    const float* Wsel = W + blockIdx.z * DD;      // W rows have length 2*DD
    float* outp = (blockIdx.z == 0) ? Wh1 : Wh2;

    v8f acc = {};
#pragma unroll 4
    for (int c0 = 0; c0 < DD; c0 += 4) {
        const int ka = c0 + half * 2;
        v2f av = *(const v2f*)(H    + (size_t)(row0 + l16) * DD        + ka);
        v2f bv = *(const v2f*)(Wsel + (size_t)(col0 + l16) * (2 * DD)  + ka);
        acc = __builtin_amdgcn_wmma_f32_16x16x4_f32(
            false, av, false, bv, (short)0, acc, false, false);
    }
#pragma unroll
    for (int r = 0; r < 8; ++r) {
        const int row = row0 + r + half * 8;
        outp[(size_t)row * DD + (col0 + l16)] = acc[r];
    }
}

// ---------------------------------------------------------------------------
// Kernel 2 (dominant pass): S[i,j] = sum_k a[k]*lrelu(Wh1[i,k]+Wh2[j,k]).
// 256-thread block -> 32x32 tile of S; each thread a 2x2 register tile.
// K blocked by 64; LDS rows padded to 68 floats so the 16 tx lanes' b128
// reads cover all 64 banks conflict-free. Inner step: 5 ds_load_b128 feed
// 16 FMA (+adds/lrelu) -> VALU-bound instead of LDS-issue-bound.
// ---------------------------------------------------------------------------
__global__ void __launch_bounds__(256)
k_scores(const float* __restrict__ Wh1, const float* __restrict__ Wh2,
         const float* __restrict__ a, float* __restrict__ S) {
    __shared__ float sA[32][68];
    __shared__ float sB[32][68];
    __shared__ float sa[DD];

    const int t  = threadIdx.x;
    const int tx = t & 15;
    const int ty = t >> 4;
    const int i0 = blockIdx.x * 32;
    const int j0 = blockIdx.y * 32;

    sa[t] = a[t];

    float acc00 = 0.0f, acc01 = 0.0f, acc10 = 0.0f, acc11 = 0.0f;

    for (int kb = 0; kb < DD; kb += 64) {
        __syncthreads();   // WAR: previous block's readers done before refill
        // Stage 32x64 of Wh1 and Wh2: 512 v4 chunks each, 2 per thread.
#pragma unroll
        for (int p = 0; p < 2; ++p) {
            const int chunk = t + p * 256;
            const int r = chunk >> 4;          // 0..31
            const int c = (chunk & 15) * 4;    // 0..60
            *(v4f*)&sA[r][c] = *(const v4f*)(Wh1 + (size_t)(i0 + r) * DD + kb + c);
            *(v4f*)&sB[r][c] = *(const v4f*)(Wh2 + (size_t)(j0 + r) * DD + kb + c);
        }
        __syncthreads();

#pragma unroll
        for (int k = 0; k < 64; k += 4) {
            const v4f A0 = *(const v4f*)&sA[ty     ][k];
            const v4f A1 = *(const v4f*)&sA[ty + 16][k];
            const v4f B0 = *(const v4f*)&sB[tx     ][k];
            const v4f B1 = *(const v4f*)&sB[tx + 16][k];
            const v4f av = *(const v4f*)&sa[kb + k];
#pragma unroll
            for (int u = 0; u < 4; ++u) {
                const float w = av[u];
                acc00 = fmaf(w, lrelu(A0[u] + B0[u]), acc00);
                acc01 = fmaf(w, lrelu(A0[u] + B1[u]), acc01);
                acc10 = fmaf(w, lrelu(A1[u] + B0[u]), acc10);
                acc11 = fmaf(w, lrelu(A1[u] + B1[u]), acc11);
            }
        }
    }

    S[(size_t)(i0 + ty     ) * NN + (j0 + tx     )] = acc00;
    S[(size_t)(i0 + ty     ) * NN + (j0 + tx + 16)] = acc01;
    S[(size_t)(i0 + ty + 16) * NN + (j0 + tx     )] = acc10;
    S[(size_t)(i0 + ty + 16) * NN + (j0 + tx + 16)] = acc11;
}

// ---------------------------------------------------------------------------
// Kernel 3: per-row unnormalized softmax: S[i,:] <- exp(S[i,:] - max);
// rowsum[i] = sum. Normalization folded into the final GEMM store.
// ---------------------------------------------------------------------------
__global__ void __launch_bounds__(256)
k_softmax(float* __restrict__ S, float* __restrict__ rowsum) {
    __shared__ float red[256];
    const int i = blockIdx.x;
    const int t = threadIdx.x;
    float* row = S + (size_t)i * NN;

    float m = -INFINITY;
    for (int j = t; j < NN; j += 256) m = fmaxf(m, row[j]);
    red[t] = m; __syncthreads();
    for (int s = 128; s > 0; s >>= 1) {
        if (t < s) red[t] = fmaxf(red[t], red[t + s]);
        __syncthreads();
    }
    m = red[0]; __syncthreads();

    float sum = 0.0f;
    for (int j = t; j < NN; j += 256) {
        const float p = __expf(row[j] - m);
        row[j] = p;
        sum += p;
    }
    red[t] = sum; __syncthreads();
    for (int s = 128; s > 0; s >>= 1) {
        if (t < s) red[t] += red[t + s];
        __syncthreads();
    }
    if (t == 0) rowsum[i] = red[0];
}

// ---------------------------------------------------------------------------
// Kernel 4: out = (P @ H) * (1/rowsum) via WMMA f32 16x16x4; K = 1024.
// A (P row) K-contiguous -> v2f load; B[k][n]=H[k][n] lanes contiguous in n.
// ---------------------------------------------------------------------------
__global__ void __launch_bounds__(32)
k_out_gemm(const float* __restrict__ P, const float* __restrict__ H,
           const float* __restrict__ rowsum, float* __restrict__ out) {
    const int lane = threadIdx.x;
    const int half = lane >> 4;
    const int l16  = lane & 15;
    const int row0 = blockIdx.x * 16;
    const int col0 = blockIdx.y * 16;

    v8f acc = {};
#pragma unroll 4
    for (int c0 = 0; c0 < NN; c0 += 4) {
        const int ka = c0 + half * 2;
        v2f av = *(const v2f*)(P + (size_t)(row0 + l16) * NN + ka);
        v2f bv;
        bv.x = H[(size_t)(ka    ) * DD + (col0 + l16)];
        bv.y = H[(size_t)(ka + 1) * DD + (col0 + l16)];
        acc = __builtin_amdgcn_wmma_f32_16x16x4_f32(
            false, av, false, bv, (short)0, acc, false, false);
    }
#pragma unroll
    for (int r = 0; r < 8; ++r) {
        const int row = row0 + r + half * 8;
        const float inv = 1.0f / rowsum[row];
        out[(size_t)row * DD + (col0 + l16)] = acc[r] * inv;
    }
}

// ---------------------------------------------------------------------------
extern "C" void kernel_launch(void* const* d_in, const int* in_sizes, int n_in,
                              void* d_out, int out_size, void* d_ws, size_t ws_size,
                              hipStream_t stream) {
    const float* H = (const float*)d_in[0];   // 1024*256
    const float* W = (const float*)d_in[1];   // 256*512
    const float* a = (const float*)d_in[2];   // 256
    float* out = (float*)d_out;               // 1024*256

    float* Wh1    = (float*)d_ws;                  // 1 MB
    float* Wh2    = Wh1 + (size_t)NN * DD;         // 1 MB
    float* S      = Wh2 + (size_t)NN * DD;         // 4 MB
    float* rowsum = S   + (size_t)NN * NN;         // 4 KB

    k_gemm_wh <<<dim3(NN / 16, DD / 16, 2), 32, 0, stream>>>(H, W, Wh1, Wh2);
    k_scores  <<<dim3(NN / 32, NN / 32),   256, 0, stream>>>(Wh1, Wh2, a, S);
    k_softmax <<<NN,                        256, 0, stream>>>(S, rowsum);
    k_out_gemm<<<dim3(NN / 16, DD / 16),    32, 0, stream>>>(S, H, rowsum, out);
}